// TransitionDownLayer_6588479832435
// MI455X (gfx1250) — compile-verified
//
#include <hip/hip_runtime.h>
#include <hip/hip_bf16.h>

typedef __attribute__((ext_vector_type(2))) float v2f;
typedef __attribute__((ext_vector_type(8))) float v8f;

#define B_      16
#define N_      8192
#define CIN     64
#define COUT    128
#define NPOINT  2048
#define KNN     16
#define M_TOTAL (B_ * N_)   // 131072 rows of the GEMM
#define MTILES  4           // 16-row M tiles per block in the GEMM
#define BN_EPS  1e-5f

// ---------------------------------------------------------------------------
// 1) Farthest point sampling: one 1024-thread block per batch.
//    Each lane keeps its 8 points + running min-distance in registers.
//    Argmax: 5-step shuffle reduce per wave, then wave0 reduces 32 winners.
// ---------------------------------------------------------------------------
__global__ __launch_bounds__(1024) void fps_kernel(const float* __restrict__ xyz,
                                                   float* __restrict__ new_xyz) {
    const int b   = blockIdx.x;
    const int tid = threadIdx.x;
    const float* pts = xyz + (size_t)b * N_ * 3;

    float px[8], py[8], pz[8], dist[8];
#pragma unroll
    for (int t = 0; t < 8; ++t) {
        int i = tid + t * 1024;
        px[t]   = pts[i * 3 + 0];
        py[t]   = pts[i * 3 + 1];
        pz[t]   = pts[i * 3 + 2];
        dist[t] = 1e10f;
    }

    __shared__ float s_rv[32];
    __shared__ int   s_ri[32];
    __shared__ int   s_far;

    const int lane = tid & 31;
    const int wid  = tid >> 5;
    int far = 0;

    for (int s = 0; s < NPOINT; ++s) {
        if (tid == 0) {
            new_xyz[((size_t)b * NPOINT + s) * 3 + 0] = pts[far * 3 + 0];
            new_xyz[((size_t)b * NPOINT + s) * 3 + 1] = pts[far * 3 + 1];
            new_xyz[((size_t)b * NPOINT + s) * 3 + 2] = pts[far * 3 + 2];
        }
        const float cx = pts[far * 3 + 0];
        const float cy = pts[far * 3 + 1];
        const float cz = pts[far * 3 + 2];

        float bv = -1.0f;
        int   bi = 0;
#pragma unroll
        for (int t = 0; t < 8; ++t) {
            float dx = px[t] - cx, dy = py[t] - cy, dz = pz[t] - cz;
            float d  = dx * dx + dy * dy + dz * dz;
            d = fminf(dist[t], d);
            dist[t] = d;
            int i = tid + t * 1024;
            if (d > bv) { bv = d; bi = i; }   // strict '>' keeps lowest index on tie
        }
#pragma unroll
        for (int off = 16; off > 0; off >>= 1) {
            float ov = __shfl_xor(bv, off, 32);
            int   oi = __shfl_xor(bi, off, 32);
            if (ov > bv || (ov == bv && oi < bi)) { bv = ov; bi = oi; }
        }
        if (lane == 0) { s_rv[wid] = bv; s_ri[wid] = bi; }
        __syncthreads();
        if (wid == 0) {
            bv = s_rv[lane];
            bi = s_ri[lane];
#pragma unroll
            for (int off = 16; off > 0; off >>= 1) {
                float ov = __shfl_xor(bv, off, 32);
                int   oi = __shfl_xor(bi, off, 32);
                if (ov > bv || (ov == bv && oi < bi)) { bv = ov; bi = oi; }
            }
            if (lane == 0) s_far = bi;
        }
        __syncthreads();
        far = s_far;
    }
}

// ---------------------------------------------------------------------------
// 2) kNN: one lane per query, points tiled through LDS, sorted top-16 in regs.
// ---------------------------------------------------------------------------
__global__ __launch_bounds__(128) void knn_kernel(const float* __restrict__ xyz,
                                                  const float* __restrict__ new_xyz,
                                                  int* __restrict__ knn_idx) {
    const int b   = blockIdx.x >> 4;
    const int qc  = blockIdx.x & 15;
    const int tid = threadIdx.x;
    const int q   = qc * 128 + tid;
    const float* pts = xyz + (size_t)b * N_ * 3;

    const float qx = new_xyz[((size_t)b * NPOINT + q) * 3 + 0];
    const float qy = new_xyz[((size_t)b * NPOINT + q) * 3 + 1];
    const float qz = new_xyz[((size_t)b * NPOINT + q) * 3 + 2];

    __shared__ float4 s_pts[1024];

    float bd[KNN];
    int   bi[KNN];
#pragma unroll
    for (int t = 0; t < KNN; ++t) { bd[t] = 3.4e38f; bi[t] = 0; }

    for (int chunk = 0; chunk < N_; chunk += 1024) {
        __syncthreads();
#pragma unroll
        for (int t = 0; t < 8; ++t) {
            int i = tid + t * 128;
            int g = chunk + i;
            s_pts[i] = make_float4(pts[g * 3 + 0], pts[g * 3 + 1], pts[g * 3 + 2], 0.0f);
        }
        __syncthreads();
#pragma unroll 4
        for (int i = 0; i < 1024; ++i) {
            float4 p = s_pts[i];
            float dx = p.x - qx, dy = p.y - qy, dz = p.z - qz;
            float d  = dx * dx + dy * dy + dz * dz;
            if (d < bd[KNN - 1]) {
                bd[KNN - 1] = d;
                bi[KNN - 1] = chunk + i;
#pragma unroll
                for (int t = KNN - 1; t > 0; --t) {
                    if (bd[t] < bd[t - 1]) {
                        float tf = bd[t]; bd[t] = bd[t - 1]; bd[t - 1] = tf;
                        int   ti = bi[t]; bi[t] = bi[t - 1]; bi[t - 1] = ti;
                    }
                }
            }
        }
    }
#pragma unroll
    for (int t = 0; t < KNN; ++t)
        knn_idx[((size_t)b * NPOINT + q) * KNN + t] = bi[t];
}

// ---------------------------------------------------------------------------
// 3) GEMM h = F(131072x64) * W^T(64x128) + bias via V_WMMA_F32_16X16X4_F32.
//    256 threads (8 waves); block stages a 64x64 A slab + whole W (128x64)
//    in LDS. Wave w owns output columns [16w,16w+16) and computes 4 M-tiles
//    (64 v_wmma per wave) with its staged B operand reused across the tiles.
//    Epilogue: write h, accumulate per-channel sum/sumsq for BN batch stats.
// ---------------------------------------------------------------------------
__global__ __launch_bounds__(256) void gemm_bn_stats_kernel(
    const float* __restrict__ F, const float* __restrict__ W,
    const float* __restrict__ bias, float* __restrict__ h,
    float* __restrict__ stats /* [0:128)=sum, [128:256)=sumsq */) {
    __shared__ float sA[MTILES * 16 * 64];   // 16 KB
    __shared__ float sW[128 * 64];           // 32 KB

    const int tid = threadIdx.x;
    const int m0  = blockIdx.x * (MTILES * 16);

    // stage W (2048 float4) and the 64-row A slab (1024 float4)
    const float4* W4  = (const float4*)W;
    float4*       sW4 = (float4*)sW;
#pragma unroll
    for (int t = 0; t < 8; ++t) sW4[tid + t * 256] = W4[tid + t * 256];
    const float4* F4  = (const float4*)(F + (size_t)m0 * CIN);
    float4*       sA4 = (float4*)sA;
#pragma unroll
    for (int t = 0; t < 4; ++t) sA4[tid + t * 256] = F4[tid + t * 256];
    __syncthreads();

    const int wave = tid >> 5;
    const int lane = tid & 31;
    const int hl   = lane >> 4;      // which 16-lane half
    const int ml   = lane & 15;      // row (A) / col (B,C)
    const int n0   = wave * 16;
    const int kb   = hl * 2;         // K sub-offset per ISA f32 A/B layout

    v8f c0 = {}, c1 = {}, c2 = {}, c3 = {};
#pragma unroll
    for (int kk = 0; kk < CIN; kk += 4) {
        v2f bv = *(const v2f*)&sW[(n0 + ml) * CIN + kk + kb];
        v2f a0 = *(const v2f*)&sA[(0 * 16 + ml) * CIN + kk + kb];
        v2f a1 = *(const v2f*)&sA[(1 * 16 + ml) * CIN + kk + kb];
        v2f a2 = *(const v2f*)&sA[(2 * 16 + ml) * CIN + kk + kb];
        v2f a3 = *(const v2f*)&sA[(3 * 16 + ml) * CIN + kk + kb];
        c0 = __builtin_amdgcn_wmma_f32_16x16x4_f32(false, a0, false, bv, (short)0, c0, false, false);
        c1 = __builtin_amdgcn_wmma_f32_16x16x4_f32(false, a1, false, bv, (short)0, c1, false, false);
        c2 = __builtin_amdgcn_wmma_f32_16x16x4_f32(false, a2, false, bv, (short)0, c2, false, false);
        c3 = __builtin_amdgcn_wmma_f32_16x16x4_f32(false, a3, false, bv, (short)0, c3, false, false);
    }

    const int   n  = n0 + ml;
    const float bb = bias[n];
    float lsum = 0.0f, lsq = 0.0f;
#pragma unroll
    for (int mt = 0; mt < MTILES; ++mt) {
        v8f c = (mt == 0) ? c0 : (mt == 1) ? c1 : (mt == 2) ? c2 : c3;
#pragma unroll
        for (int v = 0; v < 8; ++v) {
            int   m   = m0 + mt * 16 + v + hl * 8;   // C/D: VGPR v -> row v (+8 upper half)
            float val = c[v] + bb;
            h[(size_t)m * COUT + n] = val;
            lsum += val;
            lsq  += val * val;
        }
    }
    atomicAdd(&stats[n], lsum);
    atomicAdd(&stats[COUT + n], lsq);
}

// ---------------------------------------------------------------------------
// 4) BN finalize: scale = gamma/sqrt(var+eps), shift = beta - mean*scale
// ---------------------------------------------------------------------------
__global__ void zero_stats_kernel(float* __restrict__ stats) {
    stats[threadIdx.x] = 0.0f;   // 256 threads: sum + sumsq
}

__global__ void bn_finalize_kernel(const float* __restrict__ gamma,
                                   const float* __restrict__ beta,
                                   float* __restrict__ stats) {
    const int   c    = threadIdx.x;      // 128
    const float inv  = 1.0f / (float)M_TOTAL;
    const float mean = stats[c] * inv;
    const float var  = stats[COUT + c] * inv - mean * mean;
    const float sc   = gamma[c] * rsqrtf(var + BN_EPS);
    stats[2 * COUT + c] = sc;                    // scale
    stats[3 * COUT + c] = beta[c] - mean * sc;   // shift
}

// ---------------------------------------------------------------------------
// 5) Gather K neighbors, fused BN affine + ReLU, max-pool over K.
//    h (67 MB) is L2-resident (192 MB L2) so gathered reads stay on-chip.
// ---------------------------------------------------------------------------
__global__ __launch_bounds__(128) void gather_maxpool_kernel(
    const float* __restrict__ h, const int* __restrict__ knn_idx,
    const float* __restrict__ stats, float* __restrict__ out_feat) {
    const int bq = blockIdx.x;          // b*NPOINT + q
    const int b  = bq / NPOINT;
    const int c  = threadIdx.x;
    const float sc = stats[2 * COUT + c];
    const float sh = stats[3 * COUT + c];
    const int* idx = knn_idx + (size_t)bq * KNN;

    float m = 0.0f;   // ReLU output is >= 0
#pragma unroll
    for (int k = 0; k < KNN; ++k) {
        int   n = idx[k];
        float v = h[((size_t)b * N_ + n) * COUT + c];
        v = fmaxf(v * sc + sh, 0.0f);
        m = fmaxf(m, v);
    }
    out_feat[(size_t)bq * COUT + c] = m;
}

// ---------------------------------------------------------------------------
extern "C" void kernel_launch(void* const* d_in, const int* in_sizes, int n_in,
                              void* d_out, int out_size, void* d_ws, size_t ws_size,
                              hipStream_t stream) {
    (void)in_sizes; (void)n_in; (void)out_size; (void)ws_size;
    const float* xyz      = (const float*)d_in[0];  // [B,N,3]
    const float* features = (const float*)d_in[1];  // [B,N,64]
    const float* W        = (const float*)d_in[2];  // [128,64]
    const float* bias     = (const float*)d_in[3];  // [128]
    const float* gamma    = (const float*)d_in[4];  // [128]
    const float* beta     = (const float*)d_in[5];  // [128]

    float* new_xyz  = (float*)d_out;                           // [B,NPOINT,3]
    float* out_feat = new_xyz + (size_t)B_ * NPOINT * 3;       // [B,NPOINT,128]

    // workspace layout
    float* h       = (float*)d_ws;                                           // 67 MB
    int*   knn_idx = (int*)((char*)d_ws + (size_t)M_TOTAL * COUT * sizeof(float));
    float* stats   = (float*)(knn_idx + (size_t)B_ * NPOINT * KNN);          // 512 f32

    zero_stats_kernel<<<1, 256, 0, stream>>>(stats);
    fps_kernel<<<B_, 1024, 0, stream>>>(xyz, new_xyz);
    knn_kernel<<<B_ * (NPOINT / 128), 128, 0, stream>>>(xyz, new_xyz, knn_idx);
    gemm_bn_stats_kernel<<<M_TOTAL / (MTILES * 16), 256, 0, stream>>>(features, W, bias, h, stats);
    bn_finalize_kernel<<<1, 128, 0, stream>>>(gamma, beta, stats);
    gather_maxpool_kernel<<<B_ * NPOINT, 128, 0, stream>>>(h, knn_idx, stats, out_feat);
}